// IcoUpSampleMaxIndexLayer_16664473108751
// MI455X (gfx1250) — compile-verified
//
#include <hip/hip_runtime.h>

typedef __attribute__((ext_vector_type(2))) float v2f;
typedef __attribute__((ext_vector_type(8))) float v8f;

#define NB     16      // batch
#define CIN    256     // input channels  (K of the GEMM)
#define COUT   128     // output channels (M of the GEMM)
#define VRAW   10242   // raw vertices    (N of the GEMM)
#define NVERT  40962   // upsampled vertices
#define KNB    7       // neighbors per vertex
#define NTILE  48      // v-columns per workgroup (3 x 16-wide WMMA tiles)
#define CPAD   260     // LDS row stride (floats); 260 % 64 == 4 -> conflict-free b64 frag reads

// ---------------------------------------------------------------------------
// y = 0   (scatter only populates ~VRAW of NVERT slots per (b,o) row)
// ---------------------------------------------------------------------------
__global__ void zero_out_kernel(float* __restrict__ y, int n) {
    int i = (blockIdx.x * blockDim.x + threadIdx.x) * 4;
    if (i + 3 < n) {
        *(float4*)(y + i) = make_float4(0.f, 0.f, 0.f, 0.f);
    } else {
        for (int j = i; j < n; ++j) y[j] = 0.f;
    }
}

// ---------------------------------------------------------------------------
// neigh[v][k] = up_neigh[down[v]][k]   (int32, lives in d_ws)
// ---------------------------------------------------------------------------
__global__ void build_neigh_kernel(const int* __restrict__ up,
                                   const int* __restrict__ down,
                                   int* __restrict__ neigh) {
    int i = blockIdx.x * blockDim.x + threadIdx.x;
    if (i < VRAW * KNB) {
        int v = i / KNB;
        int k = i - v * KNB;
        neigh[i] = up[down[v] * KNB + k];
    }
}

// ---------------------------------------------------------------------------
// Fused: h = W @ x_b + bias  (f32 WMMA 16x16x4), then pooled scatter into y.
// Block = 256 threads (8 waves); wave w owns channels o = 16w..16w+15 for a
// 48-column v-tile.  A fragments (W) preloaded to VGPRs, B fragments read
// from a transposed, padded LDS stage of the x tile.
// ---------------------------------------------------------------------------
__global__ __launch_bounds__(256, 1)
void gemm_pool_scatter_kernel(const float* __restrict__ x,
                              const float* __restrict__ W,
                              const float* __restrict__ bias,
                              const int*  __restrict__ mpi,
                              const int*  __restrict__ neigh,
                              float* __restrict__ y)
{
    __shared__ float xs[NTILE][CPAD];   // x tile, transposed: xs[n][c]

    const int b    = blockIdx.y;
    const int v0   = blockIdx.x * NTILE;
    const int tid  = threadIdx.x;
    const int wave = tid >> 5;
    const int lane = tid & 31;
    const int n16  = lane & 15;
    const int hi   = lane >> 4;         // half-wave select (K-pair / row+8)
    const int o0   = wave * 16;

    // Prefetch the pooling-index rows this block will consume after the GEMM.
    {
        const int o = tid & 127;
        const size_t row = (size_t)(b * COUT + o) * VRAW + v0 + (tid >> 7) * 24;
        __builtin_prefetch(mpi + row, 0, 1);   // -> global_prefetch_b8
    }

    // ---- stage x tile (CIN x NTILE) into LDS, transposed (zero-pad OOB cols)
    {
        const float* xb = x + (size_t)b * CIN * VRAW;
        for (int idx = tid; idx < CIN * (NTILE / 2); idx += 256) {
            const int c  = idx / (NTILE / 2);
            const int n2 = (idx - c * (NTILE / 2)) * 2;
            const int v  = v0 + n2;                  // even, VRAW even -> pair in-range together
            float e0 = 0.f, e1 = 0.f;
            if (v < VRAW) {
                const float2 f = *(const float2*)(xb + (size_t)c * VRAW + v);
                e0 = f.x; e1 = f.y;
            }
            xs[n2 + 0][c] = e0;
            xs[n2 + 1][c] = e1;
        }
    }
    __syncthreads();

    // ---- preload all A fragments for this wave's 16 W rows (reused across tiles)
    // ISA f32 16x4 A layout: lane(M=n16), VGPRs hold K = 2*hi, 2*hi+1 per 4-step.
    v2f a[CIN / 4];
    {
        const float* wrow = W + (o0 + n16) * CIN + hi * 2;
        #pragma unroll
        for (int kk = 0; kk < CIN / 4; ++kk)
            a[kk] = *(const v2f*)(wrow + kk * 4);
    }

    v8f acc[NTILE / 16] = {};

    // ---- K loop: 64 steps of 4, 3 WMMA tiles per step (fully unrolled so
    // a[] stays in VGPRs).  B frag: lane(N=n16) holds K = kb, kb+1 from LDS.
    #pragma unroll
    for (int kk = 0; kk < CIN / 4; ++kk) {
        const int kb = kk * 4 + hi * 2;
        #pragma unroll
        for (int t = 0; t < NTILE / 16; ++t) {
            const v2f bf = *(const v2f*)&xs[t * 16 + n16][kb];
            acc[t] = __builtin_amdgcn_wmma_f32_16x16x4_f32(
                false, a[kk], false, bf, (short)0, acc[t], false, false);
        }
    }

    // ---- bias + pooled scatter.  D layout: VGPR r -> rows r (lanes 0-15) and
    // r+8 (lanes 16-31), lane -> column.  o = o0 + r + 8*hi, v = v0 + 16t + n16.
    #pragma unroll
    for (int t = 0; t < NTILE / 16; ++t) {
        const int v = v0 + t * 16 + n16;
        if (v < VRAW) {
            #pragma unroll
            for (int r = 0; r < 8; ++r) {
                const int   o   = o0 + r + 8 * hi;
                const float h   = acc[t][r] + bias[o];
                const int   pk  = mpi[(size_t)(b * COUT + o) * VRAW + v];
                const int   idx = neigh[v * KNB + pk];
                y[(size_t)(b * COUT + o) * NVERT + idx] = h;
            }
        }
    }
}

// ---------------------------------------------------------------------------
extern "C" void kernel_launch(void* const* d_in, const int* in_sizes, int n_in,
                              void* d_out, int out_size, void* d_ws, size_t ws_size,
                              hipStream_t stream) {
    (void)in_sizes; (void)n_in; (void)ws_size;

    const float* x    = (const float*)d_in[0];
    const float* W    = (const float*)d_in[1];
    const float* bias = (const float*)d_in[2];
    const int*   up   = (const int*)d_in[3];
    const int*   down = (const int*)d_in[4];
    const int*   mpi  = (const int*)d_in[5];
    float* y     = (float*)d_out;
    int*   neigh = (int*)d_ws;            // VRAW*KNB int32 = ~287 KB scratch

    {   // zero-fill y
        const int n4 = (out_size + 3) / 4;
        zero_out_kernel<<<(n4 + 255) / 256, 256, 0, stream>>>(y, out_size);
    }

    build_neigh_kernel<<<(VRAW * KNB + 255) / 256, 256, 0, stream>>>(up, down, neigh);

    dim3 grid((VRAW + NTILE - 1) / NTILE, NB);
    gemm_pool_scatter_kernel<<<grid, 256, 0, stream>>>(x, W, bias, mpi, neigh, y);
}